// RoiPooling_56590489092219
// MI455X (gfx1250) — compile-verified
//
#include <hip/hip_runtime.h>
#include <stdint.h>

// ROI bilinear pooling (ROI-align style), CDNA5/gfx1250.
//
// feature_map: (1, 128, 128, 512) f32 NHWC   -> d_in[0]
// rois:        (1, 300, 4)        f32 (x,y,w,h ints stored as floats) -> d_in[1]
// out:         (1, 300, 7, 7, 512) f32       -> d_out
//
// Memory-bound gather+lerp (AI ~ 0.3 FLOP/B). Strategy:
//  - 1 block = 1 (roi, py) row; 128 threads, each owns 4 channels (float4).
//  - Per px cell: 4 corner float4's per thread staged into LDS via
//    global_load_async_to_lds_b128 (ASYNCcnt), double-buffered so cell px+1's
//    DMA overlaps cell px's blend. s_wait_asynccnt 4 = previous stage done
//    (async loads complete in order).
//  - Each lane reads back only its own staged bytes -> no barriers needed.

#define POOL      7
#define NUM_ROIS  300
#define FM_W      128
#define CCH       512
#define TPB       128   // threads per block; TPB*4 == CCH channels

__device__ __forceinline__ void async_ld_b128(uint32_t lds_byte_off, const float* gaddr) {
  // VDST = VGPR holding workgroup-relative LDS byte address,
  // VADDR = 64-bit global address VGPR pair, saddr = off.
  asm volatile("global_load_async_to_lds_b128 %0, %1, off"
               :
               : "v"(lds_byte_off), "v"(gaddr)
               : "memory");
}

#define WAIT_ASYNCCNT(n) asm volatile("s_wait_asynccnt " #n ::: "memory")

__global__ __launch_bounds__(TPB) void roi_bilinear_kernel(
    const float* __restrict__ fm,    // (128,128,512)
    const float* __restrict__ rois,  // (300,4)
    float* __restrict__ out)         // (300,7,7,512)
{
  // [stage][corner][thread] float4 staging: 2 * 4 * 128 * 16B = 16 KB LDS.
  __shared__ float4 buf[2][4][TPB];

  const int blk = blockIdx.x;       // 0 .. NUM_ROIS*POOL-1
  const int roi = blk / POOL;
  const int py  = blk % POOL;
  const int t   = threadIdx.x;

  const float4 r = reinterpret_cast<const float4*>(rois)[roi];
  const int rx = (int)r.x;
  const int ry = (int)r.y;
  const int rw = (int)r.z;
  const int rh = (int)r.w;

  // Match reference order of ops: scale = h/POOL (f32), then py*scale.
  const float hs = (float)rh / (float)POOL;
  const float ws = (float)rw / (float)POOL;

  const float in_y = (float)py * hs;
  const int   y0   = (int)floorf(in_y);
  const int   y1   = min(y0 + 1, rh - 1);
  const float wy   = in_y - (float)y0;

  const float* row0 = fm + (size_t)(ry + y0) * (FM_W * CCH);
  const float* row1 = fm + (size_t)(ry + y1) * (FM_W * CCH);

  const int cOff = t * 4;           // this thread's channel offset (floats)

  int   gx0[POOL], gx1[POOL];
  float wxa[POOL];
#pragma unroll
  for (int px = 0; px < POOL; ++px) {
    const float in_x = (float)px * ws;
    const int   x0   = (int)floorf(in_x);
    const int   x1   = min(x0 + 1, rw - 1);
    gx0[px] = rx + x0;
    gx1[px] = rx + x1;
    wxa[px] = in_x - (float)x0;
  }

  // Low 32 bits of a generic pointer into LDS == workgroup-relative LDS addr.
  const uint32_t lbase = (uint32_t)(uintptr_t)(&buf[0][0][t]);

#define ISSUE(px_, s_)                                                          \
  do {                                                                          \
    const uint32_t l_ = lbase + (uint32_t)((s_) * 4 * TPB * 16);                \
    async_ld_b128(l_ + 0 * TPB * 16, row0 + (size_t)gx0[(px_)] * CCH + cOff);   \
    async_ld_b128(l_ + 1 * TPB * 16, row0 + (size_t)gx1[(px_)] * CCH + cOff);   \
    async_ld_b128(l_ + 2 * TPB * 16, row1 + (size_t)gx0[(px_)] * CCH + cOff);   \
    async_ld_b128(l_ + 3 * TPB * 16, row1 + (size_t)gx1[(px_)] * CCH + cOff);   \
  } while (0)

  ISSUE(0, 0);   // prime the pipeline

  float* obase = out + (size_t)(roi * POOL + py) * POOL * CCH + cOff;

#pragma unroll
  for (int px = 0; px < POOL; ++px) {
    const int s = px & 1;
    if (px + 1 < POOL) {
      ISSUE(px + 1, s ^ 1);     // overlap next cell's DMA with this blend
      WAIT_ASYNCCNT(4);         // wait until stage px's 4 loads are in LDS
    } else {
      WAIT_ASYNCCNT(0);
    }

    const float4 tl = buf[s][0][t];
    const float4 tr = buf[s][1][t];
    const float4 bl = buf[s][2][t];
    const float4 br = buf[s][3][t];
    const float  wx = wxa[px];

    float4 o;
    {
      const float top = tl.x + (tr.x - tl.x) * wx;
      const float bot = bl.x + (br.x - bl.x) * wx;
      o.x = top + (bot - top) * wy;
    }
    {
      const float top = tl.y + (tr.y - tl.y) * wx;
      const float bot = bl.y + (br.y - bl.y) * wx;
      o.y = top + (bot - top) * wy;
    }
    {
      const float top = tl.z + (tr.z - tl.z) * wx;
      const float bot = bl.z + (br.z - bl.z) * wx;
      o.z = top + (bot - top) * wy;
    }
    {
      const float top = tl.w + (tr.w - tl.w) * wx;
      const float bot = bl.w + (br.w - bl.w) * wx;
      o.w = top + (bot - top) * wy;
    }

    reinterpret_cast<float4*>(obase + (size_t)px * CCH)[0] = o;
  }
#undef ISSUE
}

extern "C" void kernel_launch(void* const* d_in, const int* in_sizes, int n_in,
                              void* d_out, int out_size, void* d_ws, size_t ws_size,
                              hipStream_t stream) {
  (void)in_sizes; (void)n_in; (void)out_size; (void)d_ws; (void)ws_size;
  const float* fm   = (const float*)d_in[0];   // 1*128*128*512 f32
  const float* rois = (const float*)d_in[1];   // 1*300*4 f32
  float* out = (float*)d_out;                  // 1*300*7*7*512 f32

  roi_bilinear_kernel<<<dim3(NUM_ROIS * POOL), dim3(TPB), 0, stream>>>(fm, rois, out);
}